// QNN_67396626809314
// MI455X (gfx1250) — compile-verified
//
#include <hip/hip_runtime.h>

typedef float v2f __attribute__((ext_vector_type(2)));
typedef float v8f __attribute__((ext_vector_type(8)));

#define NCLS 6
#define WAVES_PER_BLOCK 8

// ---------------------------------------------------------------------------
// Setup kernel: one wave.
//   threads 0..15  : build Umat[k*16+n] = U[n][k]  (WMMA-1 B operand, row-major)
//                    where U = RY(L3) * CNOTring * RY(L2) * CNOTring * RY(L1)
//   threads 16..31 : build Gmat[n*16+c] = sum_w sign_w(n) * W[c][w] (zero-padded)
// ---------------------------------------------------------------------------
__global__ void qnn_setup(const float* __restrict__ qw,   // (3,4)
                          const float* __restrict__ W,    // (6,4)
                          float* __restrict__ Umat,       // 256 floats
                          float* __restrict__ Gmat)       // 256 floats
{
    int t = threadIdx.x;
    if (t < 16) {
        float st[16];
        for (int i = 0; i < 16; ++i) st[i] = (i == t) ? 1.f : 0.f;
        for (int layer = 0; layer < 3; ++layer) {
            for (int w = 0; w < 4; ++w) {
                float hh = 0.5f * qw[layer * 4 + w];
                float c = __cosf(hh), s = __sinf(hh);
                int bit = 1 << (3 - w);   // qubit w is bit (3-w) of the flat index
                for (int base = 0; base < 16; ++base) {
                    if (base & bit) continue;
                    float a0 = st[base], a1 = st[base | bit];
                    st[base]       = c * a0 - s * a1;
                    st[base | bit] = s * a0 + c * a1;
                }
            }
            if (layer < 2) {
                const int ctrl[4] = {0, 1, 2, 3};
                const int targ[4] = {1, 2, 3, 0};
                for (int g = 0; g < 4; ++g) {
                    int bc = 1 << (3 - ctrl[g]);
                    int bt = 1 << (3 - targ[g]);
                    float tmp[16];
                    for (int i = 0; i < 16; ++i)
                        tmp[i] = (i & bc) ? st[i ^ bt] : st[i];
                    for (int i = 0; i < 16; ++i) st[i] = tmp[i];
                }
            }
        }
        // column t of U: st[n] = U[n][t]  ->  B1[t][n]
        for (int n = 0; n < 16; ++n) Umat[t * 16 + n] = st[n];
    } else {
        int n = t - 16;                   // basis-state index (row of G)
        for (int c = 0; c < 16; ++c) {
            float g = 0.f;
            if (c < NCLS) {
                for (int w = 0; w < 4; ++w) {
                    float sgn = ((n >> (3 - w)) & 1) ? -1.f : 1.f;
                    g += sgn * W[c * 4 + w];
                }
            }
            Gmat[n * 16 + c] = g;
        }
    }
}

// ---------------------------------------------------------------------------
// Main kernel: one wave processes 16 samples per tile (grid-stride over tiles).
//   GEMM1: final_amps (16x16) = init_amps (16x16) @ U^T   -- 4x WMMA f32 16x16x4
//   GEMM2: logits (16x6)      = amps^2 (16x16)  @ G + b   -- 4x WMMA f32 16x16x4
//   Epilogue: logits -> LDS transpose -> per-lane (sample-major) softmax.
// ---------------------------------------------------------------------------
__global__ __launch_bounds__(256) void qnn_main(
    const float* __restrict__ x,      // (B,4)
    const float* __restrict__ bvec,   // (6,)
    const float* __restrict__ Umat,   // 16x16
    const float* __restrict__ Gmat,   // 16x16
    float* __restrict__ out,          // (B,6)
    int B, int ntiles)
{
    // per wave: [0..255] probs transpose slab, [256..511] logits slab
    __shared__ float lds[WAVES_PER_BLOCK][512];

    const int lane = threadIdx.x & 31;
    const int warp = threadIdx.x >> 5;
    const int h    = lane >> 4;       // lane half (K split for A/B operands)
    const int col  = lane & 15;
    const int gwav = blockIdx.x * WAVES_PER_BLOCK + warp;
    const int nwav = gridDim.x * WAVES_PER_BLOCK;

    // ---- hoist constant WMMA B operands + bias C into registers ----
    v2f bU[4], bG[4];
#pragma unroll
    for (int j = 0; j < 4; ++j) {
        int row = 4 * j + 2 * h;
        bU[j].x = Umat[row * 16 + col];
        bU[j].y = Umat[(row + 1) * 16 + col];
        bG[j].x = Gmat[row * 16 + col];
        bG[j].y = Gmat[(row + 1) * 16 + col];
    }
    float bcol = (col < NCLS) ? bvec[col] : 0.f;
    v8f cini;
#pragma unroll
    for (int r = 0; r < 8; ++r) cini[r] = bcol;

    for (int tile = gwav; tile < ntiles; tile += nwav) {
        // ---- per-lane sample: rank-1 init amplitudes from trig products ----
        int s = tile * 16 + col;
        int sc = (s < B) ? s : (B - 1);           // clamp: keep EXEC full
        float4 xa = ((const float4*)x)[sc];
        float c0, s0, c1, s1, c2, s2, c3, s3;
        __sincosf(0.5f * xa.x, &s0, &c0);
        __sincosf(0.5f * xa.y, &s1, &c1);
        __sincosf(0.5f * xa.z, &s2, &c2);
        __sincosf(0.5f * xa.w, &s3, &c3);
        float g2 = h ? s2 : c2;                   // bit1 of amp index = lane half
        v2f a[4];
#pragma unroll
        for (int j = 0; j < 4; ++j) {             // bits3:2 of amp index = j
            float f0 = (j >> 1) ? s0 : c0;
            float f1 = (j & 1)  ? s1 : c1;
            float pre = f0 * f1 * g2;
            a[j].x = pre * c3;                    // bit0 = 0
            a[j].y = pre * s3;                    // bit0 = 1
        }

        // ---- GEMM1: final amplitudes ----
        v8f acc = {};
#pragma unroll
        for (int j = 0; j < 4; ++j)
            acc = __builtin_amdgcn_wmma_f32_16x16x4_f32(
                false, a[j], false, bU[j], (short)0, acc, false, false);

        // ---- probabilities ----
        v8f p;
#pragma unroll
        for (int r = 0; r < 8; ++r) p[r] = acc[r] * acc[r];

        // ---- transpose C/D layout -> A layout via per-wave LDS slab ----
#pragma unroll
        for (int r = 0; r < 8; ++r)
            lds[warp][(r + 8 * h) * 16 + col] = p[r];
        asm volatile("s_wait_dscnt 0x0" ::: "memory");  // wave-local cross-lane sync
        v2f a2[4];
#pragma unroll
        for (int j = 0; j < 4; ++j) {
            int k = 4 * j + 2 * h;
            a2[j].x = lds[warp][col * 16 + k];
            a2[j].y = lds[warp][col * 16 + k + 1];
        }

        // ---- GEMM2: logits = probs @ G + b ----
        v8f acc2 = cini;
#pragma unroll
        for (int j = 0; j < 4; ++j)
            acc2 = __builtin_amdgcn_wmma_f32_16x16x4_f32(
                false, a2[j], false, bG[j], (short)0, acc2, false, false);

        // ---- transpose logits to sample-major via second LDS slab ----
#pragma unroll
        for (int r = 0; r < 8; ++r)
            lds[warp][256 + (r + 8 * h) * 16 + col] = acc2[r];
        asm volatile("s_wait_dscnt 0x0" ::: "memory");

        // ---- in-lane softmax: lanes 0..15 each own one sample ----
        int ms = tile * 16 + lane;                // sample of this lane
        if (lane < 16 && ms < B) {
            const float* Lp = &lds[warp][256 + lane * 16];
            float4 z03 = *(const float4*)(Lp);    // classes 0..3 (b128)
            float2 z45 = *(const float2*)(Lp + 4);// classes 4..5 (b64)
            float z[NCLS] = {z03.x, z03.y, z03.z, z03.w, z45.x, z45.y};
            float m = z[0];
#pragma unroll
            for (int c = 1; c < NCLS; ++c) m = fmaxf(m, z[c]);
            float e[NCLS], sum = 0.f;
#pragma unroll
            for (int c = 0; c < NCLS; ++c) { e[c] = __expf(z[c] - m); sum += e[c]; }
            float inv = __frcp_rn(sum);
            float* op = out + (size_t)ms * NCLS;  // 24B per sample, 8B aligned
            ((float2*)op)[0] = make_float2(e[0] * inv, e[1] * inv);
            ((float2*)op)[1] = make_float2(e[2] * inv, e[3] * inv);
            ((float2*)op)[2] = make_float2(e[4] * inv, e[5] * inv);
        }
    }
}

// ---------------------------------------------------------------------------
extern "C" void kernel_launch(void* const* d_in, const int* in_sizes, int n_in,
                              void* d_out, int out_size, void* d_ws, size_t ws_size,
                              hipStream_t stream) {
    const float* x  = (const float*)d_in[0];   // (B,4)
    const float* qw = (const float*)d_in[1];   // (3,4)
    const float* W  = (const float*)d_in[2];   // (6,4)
    const float* b  = (const float*)d_in[3];   // (6,)
    float* out = (float*)d_out;

    int B = in_sizes[0] / 4;
    float* Umat = (float*)d_ws;
    float* Gmat = Umat + 256;

    qnn_setup<<<1, 32, 0, stream>>>(qw, W, Umat, Gmat);

    int ntiles = (B + 15) / 16;
    int blocksNeeded = (ntiles + WAVES_PER_BLOCK - 1) / WAVES_PER_BLOCK;
    int blocks = blocksNeeded < 2048 ? blocksNeeded : 2048;
    if (blocks < 1) blocks = 1;
    qnn_main<<<blocks, 256, 0, stream>>>(x, b, Umat, Gmat, out, B, ntiles);
}